// GNNModel_62904091017570
// MI455X (gfx1250) — compile-verified
//
#include <hip/hip_runtime.h>

typedef __attribute__((ext_vector_type(2))) float v2f;
typedef __attribute__((ext_vector_type(8))) float v8f;

// ---------------- elementwise / graph kernels ----------------

__global__ void k_fill(float* __restrict__ p, float v, int n) {
    int i = blockIdx.x * blockDim.x + threadIdx.x;
    if (i < n) p[i] = v;
}

__global__ void k_deg_edges(const int* __restrict__ dst, float* __restrict__ deg, int E) {
    int i = blockIdx.x * blockDim.x + threadIdx.x;
    if (i < E) atomicAdd(&deg[dst[i]], 1.0f);
}

__global__ void k_rsqrt(const float* __restrict__ deg, float* __restrict__ dinv, int n) {
    int i = blockIdx.x * blockDim.x + threadIdx.x;
    if (i < n) {
        float d = deg[i];
        dinv[i] = d > 0.0f ? rsqrtf(d) : 0.0f;
    }
}

// out[i,:] = h[i,:] * dinv[i]^2   (self-loop term; also initializes the accumulator)
__global__ void k_selfloop(const float4* __restrict__ h, const float* __restrict__ dinv,
                           float4* __restrict__ out, int n4) {
    int i = blockIdx.x * blockDim.x + threadIdx.x;
    if (i < n4) {
        int node = i >> 5;              // 32 float4 per 128-ch row
        float s = dinv[node];
        s = s * s;
        float4 v = h[i];
        v.x *= s; v.y *= s; v.z *= s; v.w *= s;
        out[i] = v;
    }
}

// one wave per edge: out[dst,:] += h[src,:] * dinv[src]*dinv[dst]
__global__ void k_edge_agg(const int* __restrict__ src, const int* __restrict__ dst,
                           const float* __restrict__ dinv, const float* __restrict__ h,
                           float* __restrict__ out, int E) {
    int e = blockIdx.x * (blockDim.x >> 5) + (threadIdx.x >> 5);
    if (e >= E) return;
    int lane = threadIdx.x & 31;
    int s = src[e], d = dst[e];
    float norm = dinv[s] * dinv[d];
    const float4* hs = (const float4*)(h + (size_t)s * 128);
    float* od = out + (size_t)d * 128;
    float4 v = hs[lane];                // coalesced 128B per quarter-wave
    int c = lane * 4;
    atomicAdd(&od[c + 0], v.x * norm);
    atomicAdd(&od[c + 1], v.y * norm);
    atomicAdd(&od[c + 2], v.z * norm);
    atomicAdd(&od[c + 3], v.w * norm);
}

__global__ void k_bias_relu(float4* __restrict__ x, const float4* __restrict__ b, int n4) {
    int i = blockIdx.x * blockDim.x + threadIdx.x;
    if (i < n4) {
        float4 v = x[i];
        float4 bb = b[i & 31];
        v.x = fmaxf(v.x + bb.x, 0.0f);
        v.y = fmaxf(v.y + bb.y, 0.0f);
        v.z = fmaxf(v.z + bb.z, 0.0f);
        v.w = fmaxf(v.w + bb.w, 0.0f);
        x[i] = v;
    }
}

__global__ void k_pool_accum(const float* __restrict__ h, const int* __restrict__ batch,
                             float* __restrict__ psum, float* __restrict__ cnt, int N) {
    int i = blockIdx.x * blockDim.x + threadIdx.x;   // N*32 threads
    if (i >= N * 32) return;
    int node = i >> 5, q = i & 31;
    int g = batch[node];
    float4 v = ((const float4*)(h + (size_t)node * 128))[q];
    float* ps = psum + (size_t)g * 128 + q * 4;
    atomicAdd(ps + 0, v.x);
    atomicAdd(ps + 1, v.y);
    atomicAdd(ps + 2, v.z);
    atomicAdd(ps + 3, v.w);
    if (q == 0) atomicAdd(&cnt[g], 1.0f);
}

__global__ void k_pool_div(const float* __restrict__ psum, const float* __restrict__ cnt,
                           float* __restrict__ pooled, int n) {
    int i = blockIdx.x * blockDim.x + threadIdx.x;
    if (i < n) {
        int g = i >> 7;
        pooled[i] = psum[i] / fmaxf(cnt[g], 1.0f);
    }
}

// ---------------- WMMA f32 GEMM:  out[N,KOUT] = A[N,128] x W[128,KOUT] (+bias)(relu) ----
// grid.x covers rows (128 per block = 8 waves x 16 rows), grid.y covers 64-col slabs.
// Weight slab is stored in LDS row-pair-interleaved so each B fragment
// {W[k+kh][col], W[k+kh+1][col]} is a single aligned ds_load_b64.
template <int KOUT>
__global__ __launch_bounds__(256) void k_gemm_wmma(const float* __restrict__ A,
                                                   const float* __restrict__ W,
                                                   const float* __restrict__ bias,
                                                   float* __restrict__ out, int N, int relu) {
    constexpr int K = 128;
    __shared__ v2f ldsW[(K / 2) * 64];   // 32 KB: [row-pair j][col] -> (W[2j][c], W[2j+1][c])

    const int cb = blockIdx.y * 64;      // column base of this slab

    // cooperative slab load: each thread fills row-pairs, one ds_store_b64 each
    for (int i = threadIdx.x; i < (K / 2) * 64; i += 256) {
        int j = i >> 6, c = i & 63;      // j = row pair, c = column within slab
        v2f p;
        p.x = W[(size_t)(2 * j) * KOUT + cb + c];
        p.y = W[(size_t)(2 * j + 1) * KOUT + cb + c];
        ldsW[i] = p;
    }
    __syncthreads();

    const int wave = threadIdx.x >> 5;
    const int lane = threadIdx.x & 31;
    const int lh = lane & 15;                 // M (for A) / N (for B,C) within tile
    const int kh = (lane >> 4) << 1;          // 0 or 2: K sub-offset for upper half-wave
    const int rowBase = blockIdx.x * 128 + wave * 16;
    if (rowBase >= N) return;                 // uniform per wave: EXEC stays all-ones

    const int arow = (rowBase + lh < N) ? (rowBase + lh) : (N - 1);
    const float* Arow = A + (size_t)arow * K;

    v8f acc[4] = {};
    for (int k = 0; k < K; k += 4) {
        // A 16x4 f32 fragment: VGPR v holds K = k + kh + v, row M = lh
        v2f a = *(const v2f*)(Arow + k + kh);
        const int j = (k + kh) >> 1;          // row-pair index in LDS
#pragma unroll
        for (int t = 0; t < 4; ++t) {
            // B 4x16 f32 fragment: one aligned b64, lands in an even VGPR pair
            v2f b = ldsW[j * 64 + t * 16 + lh];
            acc[t] = __builtin_amdgcn_wmma_f32_16x16x4_f32(
                false, a, false, b, (short)0, acc[t], false, false);
        }
    }

    const int mOff = (lane < 16) ? 0 : 8;     // C/D layout: lanes 16-31 hold M = v+8
#pragma unroll
    for (int t = 0; t < 4; ++t) {
        int col = cb + t * 16 + lh;
        float bv = bias ? bias[col] : 0.0f;
#pragma unroll
        for (int v = 0; v < 8; ++v) {
            int row = rowBase + mOff + v;
            if (row < N) {
                float val = acc[t][v] + bv;
                if (relu) val = fmaxf(val, 0.0f);
                out[(size_t)row * KOUT + col] = val;
            }
        }
    }
}

// ---------------- launcher ----------------

extern "C" void kernel_launch(void* const* d_in, const int* in_sizes, int n_in,
                              void* d_out, int out_size, void* d_ws, size_t ws_size,
                              hipStream_t stream) {
    const float* x  = (const float*)d_in[0];
    const int* ei   = (const int*)d_in[1];
    const int* batch = (const int*)d_in[2];
    const float* W1 = (const float*)d_in[3];
    const float* b1 = (const float*)d_in[4];
    const float* W2 = (const float*)d_in[5];
    const float* b2 = (const float*)d_in[6];
    const float* W3 = (const float*)d_in[7];
    const float* b3 = (const float*)d_in[8];
    const float* W4 = (const float*)d_in[9];
    const float* b4 = (const float*)d_in[10];
    float* outp = (float*)d_out;

    const int N = in_sizes[0] / 128;     // 100000 nodes
    const int E = in_sizes[1] / 2;       // 1.6M edges
    const int G = out_size / 64;         // 512 graphs

    const int* src = ei;
    const int* dst = ei + E;

    float* ws = (float*)d_ws;
    float* deg    = ws;                                  // N
    float* dinv   = deg + N;                             // N
    float* buf0   = dinv + N;                            // N*128  (h = A@W)
    float* buf1   = buf0 + (size_t)N * 128;              // N*128  (aggregated)
    float* psum   = buf1 + (size_t)N * 128;              // G*128
    float* cnt    = psum + (size_t)G * 128;              // G
    float* pooled = cnt + G;                             // G*128
    float* tbuf   = pooled + (size_t)G * 128;            // G*128

    auto cdiv = [](int a, int b) { return (a + b - 1) / b; };
    const int n4 = N * 32;               // float4 count of a [N,128] buffer

    // symmetric-norm degrees (self-loop contributes 1)
    k_fill<<<cdiv(N, 256), 256, 0, stream>>>(deg, 1.0f, N);
    k_deg_edges<<<cdiv(E, 256), 256, 0, stream>>>(dst, deg, E);
    k_rsqrt<<<cdiv(N, 256), 256, 0, stream>>>(deg, dinv, N);

    // ---- conv layer 1 ----
    k_gemm_wmma<128><<<dim3(cdiv(N, 128), 2), 256, 0, stream>>>(x, W1, nullptr, buf0, N, 0);
    k_selfloop<<<cdiv(n4, 256), 256, 0, stream>>>((const float4*)buf0, dinv, (float4*)buf1, n4);
    k_edge_agg<<<cdiv(E, 8), 256, 0, stream>>>(src, dst, dinv, buf0, buf1, E);
    k_bias_relu<<<cdiv(n4, 256), 256, 0, stream>>>((float4*)buf1, (const float4*)b1, n4);

    // ---- conv layer 2 ----
    k_gemm_wmma<128><<<dim3(cdiv(N, 128), 2), 256, 0, stream>>>(buf1, W2, nullptr, buf0, N, 0);
    k_selfloop<<<cdiv(n4, 256), 256, 0, stream>>>((const float4*)buf0, dinv, (float4*)buf1, n4);
    k_edge_agg<<<cdiv(E, 8), 256, 0, stream>>>(src, dst, dinv, buf0, buf1, E);
    k_bias_relu<<<cdiv(n4, 256), 256, 0, stream>>>((float4*)buf1, (const float4*)b2, n4);

    // ---- global mean pool ----
    k_fill<<<cdiv(G * 128, 256), 256, 0, stream>>>(psum, 0.0f, G * 128);
    k_fill<<<cdiv(G, 256), 256, 0, stream>>>(cnt, 0.0f, G);
    k_pool_accum<<<cdiv(N * 32, 256), 256, 0, stream>>>(buf1, batch, psum, cnt, N);
    k_pool_div<<<cdiv(G * 128, 256), 256, 0, stream>>>(psum, cnt, pooled, G * 128);

    // ---- MLP head ----
    k_gemm_wmma<128><<<dim3(cdiv(G, 128), 2), 256, 0, stream>>>(pooled, W3, b3, tbuf, G, 1);
    k_gemm_wmma<64><<<dim3(cdiv(G, 128), 1), 256, 0, stream>>>(tbuf, W4, b4, outp, G, 0);
}